// GraphConv_66563403153748
// MI455X (gfx1250) — compile-verified
//
#include <hip/hip_runtime.h>
#include <cstdint>
#include <cstddef>

#define DIM 128

// CDNA5 async-load-to-LDS path (gfx1250): stage the edge stream through the
// async engine (ASYNCcnt) instead of the normal vector-load path.
#if defined(__AMDGCN__) &&                                          \
    __has_builtin(__builtin_amdgcn_global_load_async_to_lds_b32) && \
    __has_builtin(__builtin_amdgcn_s_wait_asynccnt)
#define USE_ASYNC 1
#else
#define USE_ASYNC 0
#endif

#define GLOBAL_AS_I32(p) ((__attribute__((address_space(1))) int*)(p))
#define LDS_AS_I32(p)    ((__attribute__((address_space(3))) int*)(p))

typedef float f4 __attribute__((ext_vector_type(4)));

__device__ __forceinline__ void gatomic_add(float* p, float v) {
  // fp32 global atomic add, result unused -> non-returning form (STOREcnt,
  // no wait on the critical path).
  (void)__hip_atomic_fetch_add(p, v, __ATOMIC_RELAXED, __HIP_MEMORY_SCOPE_AGENT);
}

__global__ __launch_bounds__(256) void spmm_coo_scatter(
    const float* __restrict__ user_emb,   // [n_users, 128]
    const float* __restrict__ item_emb,   // [n_items, 128]
    const float* __restrict__ mat_val,    // [nnz]
    const int*   __restrict__ mat_row,    // [nnz]
    const int*   __restrict__ mat_col,    // [nnz]
    float* __restrict__ user_agg,         // [n_users, 128]
    float* __restrict__ item_agg,         // [n_items, 128]
    int nnz)
{
  const int tid  = threadIdx.x;
  const int lane = tid & 31;
  const int warp = tid >> 5;
  const int base = blockIdx.x * 256;

#if USE_ASYNC
  __shared__ float s_val[256];
  __shared__ int   s_row[256];
  __shared__ int   s_col[256];
  {
    int e  = base + tid;
    int ec = e < nnz ? e : nnz - 1;   // clamp so OOB lanes stage a safe address
    __builtin_amdgcn_global_load_async_to_lds_b32(
        GLOBAL_AS_I32(mat_val + ec), LDS_AS_I32(&s_val[tid]), 0, 0);
    __builtin_amdgcn_global_load_async_to_lds_b32(
        GLOBAL_AS_I32(mat_row + ec), LDS_AS_I32(&s_row[tid]), 0, 0);
    __builtin_amdgcn_global_load_async_to_lds_b32(
        GLOBAL_AS_I32(mat_col + ec), LDS_AS_I32(&s_col[tid]), 0, 0);
    __builtin_amdgcn_s_wait_asynccnt(0);
  }
  __syncthreads();
#else
  int e  = base + tid;
  int ec = e < nnz ? e : nnz - 1;
  float rv = __builtin_nontemporal_load(mat_val + ec);  // streaming, no reuse
  int   rr = __builtin_nontemporal_load(mat_row + ec);
  int   rc = __builtin_nontemporal_load(mat_col + ec);
#endif

  const f4* __restrict__ ue4 = (const f4*)user_emb;
  const f4* __restrict__ ie4 = (const f4*)item_emb;

  // One wave processes 32 edges; per edge each lane owns 4 consecutive floats
  // (one B128 gather per table, 4 coalesced dword atomics per direction).
#pragma unroll 2
  for (int j = 0; j < 32; ++j) {
    int ej = base + warp * 32 + j;
    if (ej >= nnz) break;           // wave-uniform exit
#if USE_ASYNC
    float vj   = s_val[warp * 32 + j];   // uniform LDS address -> broadcast
    int   rowj = s_row[warp * 32 + j];
    int   colj = s_col[warp * 32 + j];
#else
    float vj   = __shfl(rv, j, 32);
    int   rowj = __shfl(rr, j, 32);
    int   colj = __shfl(rc, j, 32);
#endif
    f4 iv = ie4[(size_t)colj * (DIM / 4) + lane];   // item_emb[col]  (L2 hot)
    f4 uv = ue4[(size_t)rowj * (DIM / 4) + lane];   // user_emb[row]  (L2 hot)

    float* up = user_agg + (size_t)rowj * DIM + lane * 4;
    float* ip = item_agg + (size_t)colj * DIM + lane * 4;

    gatomic_add(up + 0, vj * iv.x);
    gatomic_add(up + 1, vj * iv.y);
    gatomic_add(up + 2, vj * iv.z);
    gatomic_add(up + 3, vj * iv.w);

    gatomic_add(ip + 0, vj * uv.x);
    gatomic_add(ip + 1, vj * uv.y);
    gatomic_add(ip + 2, vj * uv.z);
    gatomic_add(ip + 3, vj * uv.w);
  }
}

extern "C" void kernel_launch(void* const* d_in, const int* in_sizes, int n_in,
                              void* d_out, int out_size, void* d_ws, size_t ws_size,
                              hipStream_t stream) {
  const float* user_emb = (const float*)d_in[0];
  const float* item_emb = (const float*)d_in[1];
  const float* mat_val  = (const float*)d_in[2];
  const int*   mat_row  = (const int*)d_in[3];
  const int*   mat_col  = (const int*)d_in[4];

  const int nnz     = in_sizes[2];
  const int n_users = in_sizes[0] / DIM;

  float* user_agg = (float*)d_out;
  float* item_agg = user_agg + (size_t)n_users * DIM;

  // d_out is poisoned by the harness; atomics accumulate, so zero it first.
  // hipMemsetAsync is graph-capture-safe (becomes a memset node).
  (void)hipMemsetAsync(d_out, 0, (size_t)out_size * sizeof(float), stream);

  const int nblocks = (nnz + 255) / 256;   // 256 edges per block, 32 per wave
  spmm_coo_scatter<<<nblocks, 256, 0, stream>>>(
      user_emb, item_emb, mat_val, mat_row, mat_col,
      user_agg, item_agg, nnz);
}